// Dummy_40398462386329
// MI455X (gfx1250) — compile-verified
//
#include <hip/hip_runtime.h>
#include <stdint.h>

#define BATCH 32
#define H 512
#define W 512
#define HW (H * W)              // 262144 = 2^18
#define NSEG (HW + 1)
#define MAXDET 100
#define TILE 64
#define HALO 8
#define LTS (TILE + 2 * HALO)   // 80
#define ITERS_PER_LAUNCH 8
#define NUM_LAUNCH 25           // 25 * 8 = 200 iterations, exact

typedef unsigned int u32x4 __attribute__((ext_vector_type(4)));
typedef int          i32x8 __attribute__((ext_vector_type(8)));
typedef int          i32x4 __attribute__((ext_vector_type(4)));

#if defined(__has_builtin)
#if __has_builtin(__builtin_amdgcn_tensor_load_to_lds) && __has_builtin(__builtin_amdgcn_s_wait_tensorcnt)
#define USE_TDM 1
#endif
#endif
#ifndef USE_TDM
#define USE_TDM 0
#endif

__device__ __forceinline__ int imax(int a, int b) { return a > b ? a : b; }

// ---------------------------------------------------------------- init labels
// label = (pixel > 127.5) ? linear_index+1 : 0   (mask = image/255 > 0.5)
__global__ __launch_bounds__(256) void k_init(const float* __restrict__ img,
                                              int* __restrict__ lab) {
  size_t i = ((size_t)blockIdx.x * 256 + threadIdx.x) * 4;
  const float4 v = *(const float4*)(img + i);
  int pix = (int)(i & (size_t)(HW - 1));
  int4 o;
  o.x = v.x > 127.5f ? pix + 1 : 0;
  o.y = v.y > 127.5f ? pix + 2 : 0;
  o.z = v.z > 127.5f ? pix + 3 : 0;
  o.w = v.w > 127.5f ? pix + 4 : 0;
  *(int4*)(lab + i) = o;
}

// ----------------------------------------------------- tiled label propagation
// One workgroup per 64x64 interior tile; 8 Jacobi iterations in LDS with an
// 8-wide halo. Tile (incl. halo) is DMA'd into LDS via the Tensor Data Mover.
__global__ __launch_bounds__(256) void k_prop(const int* __restrict__ src,
                                              int* __restrict__ dst) {
  __shared__ int tile[2][LTS * LTS];
  const int b = blockIdx.y;
  const int tx = blockIdx.x & 7, ty = blockIdx.x >> 3;
  const int gx0 = tx * TILE, gy0 = ty * TILE;
  const int tid = threadIdx.x;

#if USE_TDM
  // Pre-zero so clamped edge tiles see zero padding (matches 'SAME' padding).
  for (int i = tid; i < LTS * LTS; i += 256) tile[0][i] = 0;
  __syncthreads();
  if (tid < 32) {  // one wave issues the TDM transfer (EXEC-independent op)
    const int xs = imax(gx0 - HALO, 0);
    const int ys = imax(gy0 - HALO, 0);
    const int xe = (gx0 + TILE + HALO) < W ? (gx0 + TILE + HALO) : W;
    const int ye = (gy0 + TILE + HALO) < H ? (gy0 + TILE + HALO) : H;
    const int tw = xe - xs, th = ye - ys;
    const int lc = xs - (gx0 - HALO), lr = ys - (gy0 - HALO);
    const uint32_t ldsAddr =
        (uint32_t)(uintptr_t)(&tile[0][lr * LTS + lc]);
    const uint64_t ga =
        (uint64_t)(uintptr_t)(src + ((size_t)b << 18) + (size_t)ys * W + xs);
    // D# group0: count=1 | lds_addr | global_addr[31:0] | ga[56:32] + type=2
    u32x4 g0 = {1u, ldsAddr, (uint32_t)ga,
                (uint32_t)((ga >> 32) & 0x01FFFFFFu) | 0x80000000u};
    // D# group1: data_size=4B (2<<16), iterate_enable (1<<19);
    // tensor_dim0=tw, tensor_dim1=th, tile_dim0=tw, tile_dim1=1, dim0_stride=W
    i32x8 g1 = {(int)((2u << 16) | (1u << 19)),
                (int)((uint32_t)(tw & 0xFFFF) << 16),
                (int)((uint32_t)(th & 0xFFFF) << 16),
                (int)((uint32_t)(tw & 0xFFFF) << 16),
                1,      // tile_dim1 = 1 (row tiles), tile_dim2 = 0
                W,      // tensor_dim0_stride (elements)
                0, 0};
    // D# group2 (iterate mode): lds_addr_increment=LTS, global_addr_increment=W,
    // iterate_count = th-1  (0 => 1 iteration)
    i32x4 g2 = {0, LTS, W, (int)(((uint32_t)(th - 1) & 0xFFFF) << 16)};
    i32x4 g3 = {0, 0, 0, 0};
    i32x8 g4 = {0, 0, 0, 0, 0, 0, 0, 0};  // extra group (clang-23 6-arg form)
    __builtin_amdgcn_tensor_load_to_lds(g0, g1, g2, g3, g4, 0);
    __builtin_amdgcn_s_wait_tensorcnt(0);
  }
  __syncthreads();
#else
  for (int i = tid; i < LTS * LTS; i += 256) {
    int r = i / LTS, c = i % LTS;
    int gy = gy0 - HALO + r, gx = gx0 - HALO + c;
    tile[0][i] = (gx >= 0 && gx < W && gy >= 0 && gy < H)
                     ? src[((size_t)b << 18) + (size_t)gy * W + gx]
                     : 0;
  }
  __syncthreads();
#endif

  for (int it = 0; it < ITERS_PER_LAUNCH; ++it) {
    const int* sp = tile[it & 1];
    int* dp = tile[(it & 1) ^ 1];
    for (int i = tid; i < (LTS - 2) * (LTS - 2); i += 256) {
      int r = i / (LTS - 2) + 1, c = i % (LTS - 2) + 1;
      const int* p0 = sp + (r - 1) * LTS + c;
      const int* p1 = p0 + LTS;
      const int* p2 = p1 + LTS;
      int ctr = p1[0];
      int m = imax(imax(p0[-1], p0[0]), p0[1]);                 // v_max3
      m = imax(m, imax(imax(p1[-1], ctr), p1[1]));
      m = imax(m, imax(imax(p2[-1], p2[0]), p2[1]));
      dp[r * LTS + c] = (ctr > 0) ? m : 0;                      // mask
    }
    __syncthreads();
  }
  // after 8 iterations result is back in tile[0]; write 64x64 interior
  for (int i = tid; i < TILE * TILE; i += 256) {
    int r = i >> 6, c = i & (TILE - 1);
    dst[((size_t)b << 18) + (size_t)(gy0 + r) * W + (gx0 + c)] =
        tile[0][(r + HALO) * LTS + (c + HALO)];
  }
}

// ------------------------------------------------------------- segment counts
__global__ __launch_bounds__(256) void k_count(const int* __restrict__ lab,
                                               int* __restrict__ counts) {
  size_t i = ((size_t)blockIdx.x * 256 + threadIdx.x) * 4;
  int b = (int)(i >> 18);
  int4 L = *(const int4*)(lab + i);
  int* c = counts + (size_t)b * NSEG;
  if (L.x > 0) atomicAdd(&c[L.x], 1);
  if (L.y > 0) atomicAdd(&c[L.y], 1);
  if (L.z > 0) atomicAdd(&c[L.z], 1);
  if (L.w > 0) atomicAdd(&c[L.w], 1);
}

// -------------------------------------------------- per-image top-100 by count
// One 1024-thread (32-wave) workgroup per image; 100 argmax passes with
// lax.top_k tie-breaking (descending count, ties -> lowest segment index).
__global__ __launch_bounds__(1024) void k_topk(int* __restrict__ counts,
                                               int* __restrict__ selLabel,
                                               int* __restrict__ selCount,
                                               int* __restrict__ sortLab,
                                               int* __restrict__ sortRank,
                                               int* __restrict__ bboxI) {
  __shared__ int redC[1024], redI[1024];
  __shared__ int labs[MAXDET];
  __shared__ int sl[MAXDET], sr[MAXDET];
  const int b = blockIdx.x;
  const int tid = threadIdx.x;
  int* cnt = counts + (size_t)b * NSEG;

  for (int k = 0; k < MAXDET; ++k) {
    int bc = 0, bi = 0x7FFFFFFF;
    for (int i = tid; i < NSEG; i += 1024) {
      int c = __hip_atomic_load(&cnt[i], __ATOMIC_RELAXED,
                                __HIP_MEMORY_SCOPE_AGENT);
      if (c > bc) { bc = c; bi = i; }  // strict > keeps lowest index on ties
    }
    redC[tid] = bc; redI[tid] = bi;
    __syncthreads();
    for (int s = 512; s > 0; s >>= 1) {
      if (tid < s) {
        int c2 = redC[tid + s], i2 = redI[tid + s];
        if (c2 > redC[tid] || (c2 == redC[tid] && i2 < redI[tid])) {
          redC[tid] = c2; redI[tid] = i2;
        }
      }
      __syncthreads();
    }
    if (tid == 0) {
      int c0 = redC[0], i0 = redI[0];
      if (c0 > 0) {
        selLabel[b * MAXDET + k] = i0;
        selCount[b * MAXDET + k] = c0;
        labs[k] = i0;
        __hip_atomic_store(&cnt[i0], 0, __ATOMIC_RELAXED,
                           __HIP_MEMORY_SCOPE_AGENT);
      } else {
        selLabel[b * MAXDET + k] = 0;
        selCount[b * MAXDET + k] = 0;
        labs[k] = 0x7FFFFFFF;  // never matches a real label
      }
    }
    __threadfence();
    __syncthreads();
  }

  if (tid == 0) {  // insertion sort (label asc) with rank tracking, 100 elems
    for (int k = 0; k < MAXDET; ++k) { sl[k] = labs[k]; sr[k] = k; }
    for (int i = 1; i < MAXDET; ++i) {
      int key = sl[i], kr = sr[i], j = i - 1;
      while (j >= 0 && sl[j] > key) { sl[j + 1] = sl[j]; sr[j + 1] = sr[j]; --j; }
      sl[j + 1] = key; sr[j + 1] = kr;
    }
    for (int k = 0; k < MAXDET; ++k) {
      sortLab[b * MAXDET + k] = sl[k];
      sortRank[b * MAXDET + k] = sr[k];
    }
  }
  if (tid < MAXDET * 4) {  // init global bbox table for this image
    int f = tid & 3;
    bboxI[b * MAXDET * 4 + tid] = (f < 2) ? 0x7FFFFFFF : (int)0x80000000;
  }
}

// -------------------------------------------------- bboxes for selected labels
__global__ __launch_bounds__(256) void k_bbox(const int* __restrict__ lab,
                                              const int* __restrict__ sortLab,
                                              const int* __restrict__ sortRank,
                                              int* __restrict__ bboxI) {
  __shared__ int sl[MAXDET], sr[MAXDET];
  __shared__ int bb[MAXDET][4];
  const int b = blockIdx.y;
  const int tid = threadIdx.x;
  if (tid < MAXDET) {
    sl[tid] = sortLab[b * MAXDET + tid];
    sr[tid] = sortRank[b * MAXDET + tid];
  }
  for (int i = tid; i < MAXDET * 4; i += 256)
    ((int*)bb)[i] = ((i & 3) < 2) ? 0x7FFFFFFF : (int)0x80000000;
  __syncthreads();

  const int base = blockIdx.x * 4096;
  const int* L = lab + ((size_t)b << 18);
  for (int t = 0; t < 16; ++t) {
    int idx = base + t * 256 + tid;
    int v = L[idx];
    if (v > 0) {
      int lo = 0, hi = MAXDET - 1, r = -1;
      while (lo <= hi) {  // binary search in sorted selected labels
        int mid = (lo + hi) >> 1;
        int s = sl[mid];
        if (s == v) { r = sr[mid]; break; }
        else if (s < v) lo = mid + 1;
        else hi = mid - 1;
      }
      if (r >= 0) {
        int x = idx & (W - 1), y = (idx >> 9) & (H - 1);
        atomicMin(&bb[r][0], x);
        atomicMin(&bb[r][1], y);
        atomicMax(&bb[r][2], x);
        atomicMax(&bb[r][3], y);
      }
    }
  }
  __syncthreads();
  for (int i = tid; i < MAXDET; i += 256) {
    if (bb[i][0] != 0x7FFFFFFF) {
      int* g = bboxI + (b * MAXDET + i) * 4;
      atomicMin(&g[0], bb[i][0]);
      atomicMin(&g[1], bb[i][1]);
      atomicMax(&g[2], bb[i][2]);
      atomicMax(&g[3], bb[i][3]);
    }
  }
}

// ---------------------------------------------------------------------- emit
__global__ __launch_bounds__(256) void k_emit(const int* __restrict__ selCount,
                                              const int* __restrict__ bboxI,
                                              float* __restrict__ out) {
  int i = blockIdx.x * 256 + threadIdx.x;
  if (i >= BATCH * MAXDET) return;
  int c = selCount[i];
  float4 o = {0.f, 0.f, 0.f, 0.f};
  if (c > 0) {
    const int* g = bboxI + i * 4;
    o.x = (float)g[0]; o.y = (float)g[1]; o.z = (float)g[2]; o.w = (float)g[3];
  }
  *(float4*)(out + (size_t)i * 4) = o;
}

extern "C" void kernel_launch(void* const* d_in, const int* in_sizes, int n_in,
                              void* d_out, int out_size, void* d_ws,
                              size_t ws_size, hipStream_t stream) {
  (void)in_sizes; (void)n_in; (void)out_size; (void)ws_size;
  const float* img = (const float*)d_in[0];
  char* ws = (char*)d_ws;
  const size_t labBytes = (size_t)BATCH * HW * 4;          // 33,554,432
  const size_t cntBytes = (size_t)BATCH * NSEG * 4;        // 33,554,560
  int* labA = (int*)(ws);
  int* labB = (int*)(ws + labBytes);
  int* counts = (int*)(ws + 2 * labBytes);
  char* p = ws + 2 * labBytes + cntBytes;
  int* selLabel = (int*)p; p += BATCH * MAXDET * 4;
  int* selCount = (int*)p; p += BATCH * MAXDET * 4;
  int* sortLab  = (int*)p; p += BATCH * MAXDET * 4;
  int* sortRank = (int*)p; p += BATCH * MAXDET * 4;
  int* bboxI    = (int*)p;
  float* out = (float*)d_out;

  k_init<<<8192, 256, 0, stream>>>(img, labA);

  for (int s = 0; s < NUM_LAUNCH; ++s) {
    const int* src = (s & 1) ? labB : labA;
    int* dst = (s & 1) ? labA : labB;
    k_prop<<<dim3(64, BATCH), 256, 0, stream>>>(src, dst);
  }
  // 25 launches, final result in labB

  (void)hipMemsetAsync(counts, 0, cntBytes, stream);
  k_count<<<8192, 256, 0, stream>>>(labB, counts);
  k_topk<<<BATCH, 1024, 0, stream>>>(counts, selLabel, selCount, sortLab,
                                     sortRank, bboxI);
  k_bbox<<<dim3(64, BATCH), 256, 0, stream>>>(labB, sortLab, sortRank, bboxI);
  k_emit<<<(BATCH * MAXDET + 255) / 256, 256, 0, stream>>>(selCount, bboxI, out);
}